// HyConv_18245021073764
// MI455X (gfx1250) — compile-verified
//
#include <hip/hip_runtime.h>
#include <hip/hip_bf16.h>

// Problem constants (match reference)
#define BB 4
#define NN 10000
#define MM 10000
#define EE 160000
#define CC 128   // C_IN == C_OUT == 128

typedef __attribute__((ext_vector_type(2))) float v2f;
typedef __attribute__((ext_vector_type(8))) float v8f;

// ---------------------------------------------------------------------------
// 0) init: deg_n = deg_e = 0, x_edge = 0, out[b,n,c] = bias[c]
// ---------------------------------------------------------------------------
__global__ void __launch_bounds__(256)
init_all(float* __restrict__ deg_n, long long n1,
         float* __restrict__ deg_e, long long n2,
         float* __restrict__ x_edge, long long n3,
         float* __restrict__ out, long long n4,
         const float* __restrict__ bias) {
  long long stride = (long long)gridDim.x * blockDim.x;
  long long t = (long long)blockIdx.x * blockDim.x + threadIdx.x;
  for (long long i = t; i < n1; i += stride) deg_n[i] = 0.0f;
  for (long long i = t; i < n2; i += stride) deg_e[i] = 0.0f;
  for (long long i = t; i < n3; i += stride) x_edge[i] = 0.0f;
  for (long long i = t; i < n4; i += stride) out[i] = bias[i & (CC - 1)];
}

// ---------------------------------------------------------------------------
// 1) degree counts via f32 atomics (reference uses segment_sum of ones)
// ---------------------------------------------------------------------------
__global__ void __launch_bounds__(256)
degrees(const int* __restrict__ H, float* __restrict__ deg_n,
        float* __restrict__ deg_e) {
  long long idx = (long long)blockIdx.x * blockDim.x + threadIdx.x;
  long long total = (long long)BB * EE;
  if (idx >= total) return;
  int b = (int)(idx / EE);
  int e = (int)(idx % EE);
  int ni = H[((size_t)b * 2 + 0) * EE + e];
  int he = H[((size_t)b * 2 + 1) * EE + e];
  atomicAdd(&deg_n[(size_t)b * NN + ni], 1.0f);
  atomicAdd(&deg_e[(size_t)b * MM + he], 1.0f);
}

// ---------------------------------------------------------------------------
// 2) xt = x @ theta  via V_WMMA_F32_16X16X4_F32 (full f32 precision)
//    One block = 16 rows of x, 8 waves cover the 128 output columns.
//    x-tile staged in LDS via async global->LDS B128 copies (ASYNCcnt),
//    row stride padded to 132 floats (528B: 16B-aligned, bank-conflict-free).
// ---------------------------------------------------------------------------
__global__ void __launch_bounds__(256)
gemm_xtheta(const float* __restrict__ x, const float* __restrict__ theta,
            float* __restrict__ xt) {
  __shared__ float sA[16][CC + 4];   // 16 x 132 floats = 8448 B

  const int ntiles = NN / 16;        // 625, exact
  int tile = blockIdx.x;             // over BB * ntiles
  int b = tile / ntiles;
  int m0 = (tile % ntiles) * 16;

  const float* xb = x + ((size_t)b * NN + m0) * CC;

  // Async stage the 16x128 f32 tile: 512 16B chunks, 2 per thread.
  #pragma unroll
  for (int it = 0; it < 2; ++it) {
    int q = threadIdx.x + it * 256;     // 0..511
    int row = q >> 5;                   // 0..15
    int col4 = q & 31;                  // 0..31 (units of 4 floats)
    unsigned lds = (unsigned)(uintptr_t)(&sA[row][col4 * 4]);
    const float* g = xb + row * CC + col4 * 4;
    asm volatile("global_load_async_to_lds_b128 %0, %1, off"
                 :: "v"(lds), "v"(g) : "memory");
  }
  asm volatile("s_wait_asynccnt 0x0" ::: "memory");
  __syncthreads();

  const int lane = threadIdx.x & 31;
  const int wave = threadIdx.x >> 5; // 0..7 -> N-slab
  const int n0 = wave * 16;
  const int half = lane >> 4;        // 0: K=0/1; 1: K=2/3
  const int lid = lane & 15;         // M index (A) / N index (B,C)

  v8f acc = {};
  #pragma unroll 4
  for (int kk = 0; kk < CC; kk += 4) {
    const int k = kk + half * 2;
    v2f a;                           // A: 16x4, lane lid = row M, VGPR0/1 = K/K+1
    a.x = sA[lid][k];
    a.y = sA[lid][k + 1];
    v2f bm;                          // B: 4x16, lane lid = col N, VGPR0/1 = K/K+1
    bm.x = theta[(size_t)k * CC + n0 + lid];
    bm.y = theta[(size_t)(k + 1) * CC + n0 + lid];
    acc = __builtin_amdgcn_wmma_f32_16x16x4_f32(
        /*neg_a=*/false, a, /*neg_b=*/false, bm,
        /*c_mod=*/(short)0, acc, /*reuse_a=*/false, /*reuse_b=*/false);
  }

  // C/D layout: VGPR r -> lanes 0-15: M=r, lanes 16-31: M=8+r; N=lid
  float* op = xt + ((size_t)b * NN + m0) * CC;
  #pragma unroll
  for (int r = 0; r < 8; ++r) {
    int m = r + half * 8;
    op[(size_t)m * CC + n0 + lid] = acc[r];
  }
}

// ---------------------------------------------------------------------------
// 3) node -> hyperedge:  x_edge[b,he,:] += xt[b,ni,:] / deg_e[b,he]
//    One wave per edge: float4 per lane -> global_load_b128 gathers,
//    4x global_atomic_add_f32 scatters.
// ---------------------------------------------------------------------------
__global__ void __launch_bounds__(256)
scatter_edge(const float* __restrict__ xt, const int* __restrict__ H,
             const float* __restrict__ deg_e, float* __restrict__ x_edge) {
  long long eg = (long long)blockIdx.x * 8 + (threadIdx.x >> 5); // over BB*EE
  int lane = threadIdx.x & 31;
  int b = (int)(eg / EE);
  int e = (int)(eg % EE);
  int ni = H[((size_t)b * 2 + 0) * EE + e];
  int he = H[((size_t)b * 2 + 1) * EE + e];
  float w = 1.0f / deg_e[(size_t)b * MM + he];
  const float4* src =
      (const float4*)(xt + ((size_t)b * NN + ni) * CC) + lane;
  float4 v = *src;
  float* dst = x_edge + ((size_t)b * MM + he) * CC + lane * 4;
  atomicAdd(dst + 0, v.x * w);
  atomicAdd(dst + 1, v.y * w);
  atomicAdd(dst + 2, v.z * w);
  atomicAdd(dst + 3, v.w * w);
}

// ---------------------------------------------------------------------------
// 4) hyperedge -> node:  out[b,ni,:] += x_edge[b,he,:] / deg_n[b,ni]
//    (out pre-initialized with bias)
// ---------------------------------------------------------------------------
__global__ void __launch_bounds__(256)
scatter_node(const float* __restrict__ x_edge, const int* __restrict__ H,
             const float* __restrict__ deg_n, float* __restrict__ out) {
  long long eg = (long long)blockIdx.x * 8 + (threadIdx.x >> 5);
  int lane = threadIdx.x & 31;
  int b = (int)(eg / EE);
  int e = (int)(eg % EE);
  int ni = H[((size_t)b * 2 + 0) * EE + e];
  int he = H[((size_t)b * 2 + 1) * EE + e];
  float w = 1.0f / deg_n[(size_t)b * NN + ni];
  const float4* src =
      (const float4*)(x_edge + ((size_t)b * MM + he) * CC) + lane;
  float4 v = *src;
  float* dst = out + ((size_t)b * NN + ni) * CC + lane * 4;
  atomicAdd(dst + 0, v.x * w);
  atomicAdd(dst + 1, v.y * w);
  atomicAdd(dst + 2, v.z * w);
  atomicAdd(dst + 3, v.w * w);
}

// ---------------------------------------------------------------------------
extern "C" void kernel_launch(void* const* d_in, const int* in_sizes, int n_in,
                              void* d_out, int out_size, void* d_ws, size_t ws_size,
                              hipStream_t stream) {
  const float* x     = (const float*)d_in[0];   // [B, N, C_IN] f32
  const int*   H     = (const int*)d_in[1];     // [B, 2, E] int
  const float* theta = (const float*)d_in[2];   // [C_IN, C_OUT] f32
  const float* bias  = (const float*)d_in[3];   // [C_OUT] f32
  float* out = (float*)d_out;                   // [B, N, C_OUT] f32

  // workspace layout
  float* ws     = (float*)d_ws;
  float* xt     = ws;                                   // B*N*CC
  float* x_edge = xt + (size_t)BB * NN * CC;            // B*M*CC
  float* deg_n  = x_edge + (size_t)BB * MM * CC;        // B*N
  float* deg_e  = deg_n + (size_t)BB * NN;              // B*M

  const long long nDegN = (long long)BB * NN;
  const long long nDegE = (long long)BB * MM;
  const long long nEdgeF = (long long)BB * MM * CC;
  const long long nOutF = (long long)BB * NN * CC;

  init_all<<<4096, 256, 0, stream>>>(deg_n, nDegN, deg_e, nDegE,
                                     x_edge, nEdgeF, out, nOutF, bias);

  const long long totE = (long long)BB * EE;
  degrees<<<(int)((totE + 255) / 256), 256, 0, stream>>>(H, deg_n, deg_e);

  gemm_xtheta<<<BB * (NN / 16), 256, 0, stream>>>(x, theta, xt);

  scatter_edge<<<(int)(totE / 8), 256, 0, stream>>>(xt, H, deg_e, x_edge);
  scatter_node<<<(int)(totE / 8), 256, 0, stream>>>(x_edge, H, deg_n, out);
}